// GATv2Conv_63273458205234
// MI455X (gfx1250) — compile-verified
//
#include <hip/hip_runtime.h>
#include <math.h>

typedef __attribute__((ext_vector_type(2))) float v2f;
typedef __attribute__((ext_vector_type(8))) float v8f;

// ---------------------------------------------------------------------------
// mish(x) = x * tanh(softplus(x))
// ---------------------------------------------------------------------------
__device__ __forceinline__ float mishf(float v) {
    float sp = (v > 15.f) ? v : log1pf(expf(v));
    return v * tanhf(sp);
}

// Sign-aware float atomic max via integer atomics (lmax must be inited to -inf)
__device__ __forceinline__ void atomicMaxF(float* addr, float v) {
    if (v >= 0.f) {
        atomicMax((int*)addr, __float_as_int(v));
    } else {
        atomicMin((unsigned int*)addr, (unsigned int)__float_as_int(v));
    }
}

// ---------------------------------------------------------------------------
// K0: initialize lmax=-inf, denom=0, out=0
// ---------------------------------------------------------------------------
__global__ void gat_init_kernel(float* __restrict__ lmax, float* __restrict__ denom,
                                float* __restrict__ out, int n4, int n64) {
    int i = blockIdx.x * blockDim.x + threadIdx.x;
    int stride = gridDim.x * blockDim.x;
    for (int j = i; j < n64; j += stride) out[j] = 0.f;
    for (int j = i; j < n4; j += stride) {
        lmax[j]  = -INFINITY;
        denom[j] = 0.f;
    }
}

// ---------------------------------------------------------------------------
// K1: node projections  sp = x@Ws + bs, rp = x@Wr + br   via fp32 WMMA 16x16x4
// One wave (32 lanes) computes a [16 rows x 64 cols] tile for both matrices.
// A-frag layout (16x4 f32): lanes 0-15 -> M=lane, K={0,1}; lanes 16-31 -> M=lane-16, K={2,3}
// B-frag layout (4x16 f32): lanes 0-15 -> N=lane, K={0,1}; lanes 16-31 -> N=lane-16, K={2,3}
// D layout (16x16 f32): VGPR i: lanes 0-15 -> M=i, N=lane; lanes 16-31 -> M=8+i, N=lane-16
// ---------------------------------------------------------------------------
__global__ void proj_wmma_kernel(const float* __restrict__ x,
                                 const float* __restrict__ Ws, const float* __restrict__ bs,
                                 const float* __restrict__ Wr, const float* __restrict__ br,
                                 float* __restrict__ sp, float* __restrict__ rp, int n) {
    const int lane = threadIdx.x & 31;
    const int wave = blockIdx.x * (blockDim.x >> 5) + (threadIdx.x >> 5);
    const int ntiles = (n + 15) >> 4;
    if (wave >= ntiles) return;

    const int row0 = wave << 4;
    const int rl   = lane & 15;          // row (A) / col (B,D) within tile
    const int koff = (lane >> 4) << 1;   // 0 for lanes 0-15, 2 for lanes 16-31
    const int mrow = (lane >> 4) << 3;   // D base row: 0 or 8
    const bool fullTile = (row0 + 16 <= n);   // wave-uniform

    int r = row0 + rl;
    if (r > n - 1) r = n - 1;

    // Hoist all 16 A fragments (K = 64 in chunks of 4) into registers.
    v2f a[16];
    const float* xr = x + (size_t)r * 64 + koff;
#pragma unroll
    for (int kb = 0; kb < 16; ++kb) {
        a[kb].x = xr[kb * 4 + 0];
        a[kb].y = xr[kb * 4 + 1];
    }

#pragma unroll
    for (int m = 0; m < 2; ++m) {
        const float* W    = m ? Wr : Ws;
        const float* bias = m ? br : bs;
        float*       outp = m ? rp : sp;
#pragma unroll
        for (int ct = 0; ct < 4; ++ct) {
            const int col = ct * 16 + rl;
            const float* wc = W + (size_t)koff * 64 + col;
            v8f acc = {};
#pragma unroll
            for (int kb = 0; kb < 16; ++kb) {
                v2f b;
                b.x = wc[(kb * 4 + 0) * 64];
                b.y = wc[(kb * 4 + 1) * 64];
                acc = __builtin_amdgcn_wmma_f32_16x16x4_f32(
                    /*neg_a=*/false, a[kb], /*neg_b=*/false, b,
                    /*c_mod=*/(short)0, acc, /*reuse_a=*/false, /*reuse_b=*/false);
            }
            const float bv = bias[col];
            float* op = outp + (size_t)(row0 + mrow) * 64 + col;
            if (fullTile) {
                // uniform fast path: no per-lane EXEC churn around the stores
#pragma unroll
                for (int i = 0; i < 8; ++i) op[(size_t)i * 64] = acc[i] + bv;
            } else {
#pragma unroll
                for (int i = 0; i < 8; ++i) {
                    if (row0 + mrow + i < n) op[(size_t)i * 64] = acc[i] + bv;
                }
            }
        }
    }
}

// ---------------------------------------------------------------------------
// K2: per (edge, head) logit = dot(mish(se+re), aw) + ab ; segment max (atomic)
// ---------------------------------------------------------------------------
__global__ void edge_logits_kernel(const float* __restrict__ sp, const float* __restrict__ rp,
                                   const int* __restrict__ senders,
                                   const int* __restrict__ receivers,
                                   const float* __restrict__ aw, const float* __restrict__ abp,
                                   float* __restrict__ logits, float* __restrict__ lmax, int E) {
    int t = blockIdx.x * blockDim.x + threadIdx.x;
    if (t >= E * 4) return;
    int e = t >> 2, h = t & 3;
    int s = senders[e], r = receivers[e];

    const float4* sv = (const float4*)(sp + (size_t)s * 64 + h * 16);
    const float4* rv = (const float4*)(rp + (size_t)r * 64 + h * 16);
    const float4* wv = (const float4*)aw;

    float acc = 0.f;
#pragma unroll
    for (int i = 0; i < 4; ++i) {
        float4 a = sv[i], b = rv[i], w = wv[i];
        acc += mishf(a.x + b.x) * w.x;
        acc += mishf(a.y + b.y) * w.y;
        acc += mishf(a.z + b.z) * w.z;
        acc += mishf(a.w + b.w) * w.w;
    }
    float logit = acc + abp[0];
    logits[t] = logit;
    atomicMaxF(lmax + (size_t)r * 4 + h, logit);
}

// ---------------------------------------------------------------------------
// K3: ex = exp(logit - lmax[r,h]); accumulate segment denominator
// ---------------------------------------------------------------------------
__global__ void edge_exp_kernel(const int* __restrict__ receivers,
                                const float* __restrict__ lmax,
                                float* __restrict__ logits, float* __restrict__ denom, int E) {
    int t = blockIdx.x * blockDim.x + threadIdx.x;
    if (t >= E * 4) return;
    int e = t >> 2, h = t & 3;
    int r = receivers[e];
    float ex = expf(logits[t] - lmax[(size_t)r * 4 + h]);
    logits[t] = ex;  // overwrite logits buffer with exp values
    atomicAdd(denom + (size_t)r * 4 + h, ex);
}

// ---------------------------------------------------------------------------
// K4: alpha = ex/denom; scatter alpha * se into out[receiver]
// ---------------------------------------------------------------------------
__global__ void edge_scatter_kernel(const float* __restrict__ sp,
                                    const int* __restrict__ senders,
                                    const int* __restrict__ receivers,
                                    const float* __restrict__ ex,
                                    const float* __restrict__ denom,
                                    float* __restrict__ out, int E) {
    int t = blockIdx.x * blockDim.x + threadIdx.x;
    if (t >= E * 4) return;
    int e = t >> 2, h = t & 3;
    int s = senders[e], r = receivers[e];
    float alpha = ex[t] / denom[(size_t)r * 4 + h];

    const float4* sv = (const float4*)(sp + (size_t)s * 64 + h * 16);
    float* op = out + (size_t)r * 64 + h * 16;
#pragma unroll
    for (int i = 0; i < 4; ++i) {
        float4 a = sv[i];
        atomicAdd(op + i * 4 + 0, alpha * a.x);
        atomicAdd(op + i * 4 + 1, alpha * a.y);
        atomicAdd(op + i * 4 + 2, alpha * a.z);
        atomicAdd(op + i * 4 + 3, alpha * a.w);
    }
}

// ---------------------------------------------------------------------------
extern "C" void kernel_launch(void* const* d_in, const int* in_sizes, int n_in,
                              void* d_out, int out_size, void* d_ws, size_t ws_size,
                              hipStream_t stream) {
    const float* x   = (const float*)d_in[0];
    const float* Ws  = (const float*)d_in[1];
    const float* bs  = (const float*)d_in[2];
    const float* Wr  = (const float*)d_in[3];
    const float* br  = (const float*)d_in[4];
    const float* aw  = (const float*)d_in[5];
    const float* ab  = (const float*)d_in[6];
    const int* senders   = (const int*)d_in[7];
    const int* receivers = (const int*)d_in[8];

    const int n = in_sizes[0] / 64;   // 100000
    const int E = in_sizes[7];        // 1600000
    float* out = (float*)d_out;

    // Workspace layout (floats): sp[n*64] | rp[n*64] | lmax[n*4] | denom[n*4] | logits[E*4]
    float* ws    = (float*)d_ws;
    float* sp    = ws;
    float* rp    = sp + (size_t)n * 64;
    float* lmax  = rp + (size_t)n * 64;
    float* denom = lmax + (size_t)n * 4;
    float* logits = denom + (size_t)n * 4;

    gat_init_kernel<<<2048, 256, 0, stream>>>(lmax, denom, out, n * 4, n * 64);

    const int ntiles = (n + 15) / 16;
    const int wavesPerBlock = 8;                         // 256 threads = 8 wave32
    const int pblocks = (ntiles + wavesPerBlock - 1) / wavesPerBlock;
    proj_wmma_kernel<<<pblocks, wavesPerBlock * 32, 0, stream>>>(x, Ws, bs, Wr, br, sp, rp, n);

    const int T = E * 4;
    const int tb = 256;
    const int gb = (T + tb - 1) / tb;
    edge_logits_kernel<<<gb, tb, 0, stream>>>(sp, rp, senders, receivers, aw, ab, logits, lmax, E);
    edge_exp_kernel<<<gb, tb, 0, stream>>>(receivers, lmax, logits, denom, E);
    edge_scatter_kernel<<<gb, tb, 0, stream>>>(sp, senders, receivers, logits, denom, out, E);
}